// GoGMCModel_51668456571193
// MI455X (gfx1250) — compile-verified
//
#include <hip/hip_runtime.h>

#define HDIM 128
#define GNUM 256
#define CNUM 10
#define EPSV 1e-5f

typedef float v2f __attribute__((ext_vector_type(2)));
typedef float v8f __attribute__((ext_vector_type(8)));

// ---------------- utility kernels ----------------

__global__ void k_fill(float* __restrict__ p, float val, int n) {
    int i = blockIdx.x * blockDim.x + threadIdx.x;
    int st = gridDim.x * blockDim.x;
    for (; i < n; i += st) p[i] = val;
}

// nan->0, +inf->1e6, -inf->-1e6 (bit-exact check, immune to fast-math)
__global__ void k_clean(const float* __restrict__ x, float* __restrict__ out, int n4) {
    int i = blockIdx.x * blockDim.x + threadIdx.x;
    int st = gridDim.x * blockDim.x;
    for (; i < n4; i += st) {
        float4 v = ((const float4*)x)[i];
        float* p = (float*)&v;
#pragma unroll
        for (int j = 0; j < 4; ++j) {
            unsigned u = __float_as_uint(p[j]);
            if ((u & 0x7f800000u) == 0x7f800000u) {
                if (u & 0x007fffffu) p[j] = 0.f;                  // NaN
                else p[j] = (u >> 31) ? -1e6f : 1e6f;             // +-Inf
            }
        }
        ((float4*)out)[i] = v;
    }
}

__global__ void k_deg_accum(const int* __restrict__ dst, float* __restrict__ deg, int E) {
    int i = blockIdx.x * blockDim.x + threadIdx.x;
    int st = gridDim.x * blockDim.x;
    for (; i < E; i += st) atomicAdd(&deg[dst[i]], 1.0f);
}

__global__ void k_rsqrt_inplace(float* __restrict__ p, int n) {
    int i = blockIdx.x * blockDim.x + threadIdx.x;
    int st = gridDim.x * blockDim.x;
    for (; i < n; i += st) p[i] = rsqrtf(p[i]);   // deg >= 1 always (self loop)
}

// ---------------- WMMA fp32 GEMM: Out[N x 128] = X[N x 128] @ W[128 x 128] ----------------
// Block = 256 threads = 8 waves, owns a 32-row M tile staged into LDS with
// global_load_async_to_lds_b128 (ASYNCcnt DMA). Wave w computes column tile
// [16w,16w+16) for BOTH 16-row halves, so each B fragment feeds two
// v_wmma_f32_16x16x4_f32 (halves global B traffic per wmma).
__global__ __launch_bounds__(256) void k_gemm_wmma(const float* __restrict__ X,
                                                   const float* __restrict__ W,
                                                   float* __restrict__ Out, int N) {
    __shared__ float tileA[32 * HDIM];           // 16 KB of the 320 KB WGP LDS
    const int m0  = blockIdx.x * 32;
    const int tid = threadIdx.x;

    // stage 32x128 A tile = 1024 float4, 4 per thread, async global->LDS
    const unsigned ldsbase = (unsigned)(unsigned long long)tileA; // low 32b = LDS offset
#pragma unroll
    for (int it = 0; it < 4; ++it) {
        int f4  = tid + it * 256;                // float4 index 0..1023
        int row = f4 >> 5;                       // 32 float4 per row
        int k4  = f4 & 31;
        if (m0 + row < N) {                      // garbage rows never stored
            unsigned loff = ldsbase + (unsigned)f4 * 16u;
            const float* gp = X + (size_t)(m0 + row) * HDIM + k4 * 4;
            asm volatile("global_load_async_to_lds_b128 %0, %1, off"
                         :: "v"(loff), "v"(gp) : "memory");
        }
    }
    asm volatile("s_wait_asynccnt 0x0" ::: "memory");
    __syncthreads();

    const int lane = tid & 31;
    const int col0 = (tid >> 5) << 4;            // wave id * 16
    const int r    = lane & 15;                  // M row (A) / N col (B,C)
    const int koff = (lane >> 4) << 1;           // 0 or 2: K sub-offset per ISA layout

    v8f acc0 = {0.f, 0.f, 0.f, 0.f, 0.f, 0.f, 0.f, 0.f};
    v8f acc1 = {0.f, 0.f, 0.f, 0.f, 0.f, 0.f, 0.f, 0.f};
    const float* arow0 = tileA + r * HDIM + koff;
    const float* arow1 = tileA + (r + 16) * HDIM + koff;
    const float* bcol  = W + (size_t)koff * HDIM + (col0 + r);

#pragma unroll 4
    for (int kk = 0; kk < HDIM / 4; ++kk) {      // 2 x 32 wmma per wave
        v2f a0, a1, b;
        a0[0] = arow0[kk * 4];
        a0[1] = arow0[kk * 4 + 1];
        a1[0] = arow1[kk * 4];
        a1[1] = arow1[kk * 4 + 1];
        b[0]  = bcol[(size_t)(kk * 4) * HDIM];
        b[1]  = bcol[(size_t)(kk * 4 + 1) * HDIM];
        acc0 = __builtin_amdgcn_wmma_f32_16x16x4_f32(false, a0, false, b,
                                                     (short)0, acc0, false, false);
        acc1 = __builtin_amdgcn_wmma_f32_16x16x4_f32(false, a1, false, b,
                                                     (short)0, acc1, false, false);
    }

    // C layout: VGPR v -> row (lane>>4)*8 + v, col = col0 + (lane&15)
    const int crow = ((lane >> 4) << 3);
    const int ccol = col0 + r;
    if (m0 + 32 <= N) {                          // full-tile fast path (always, N%32-safe grid)
#pragma unroll
        for (int v = 0; v < 8; ++v) {
            Out[(size_t)(m0 + crow + v) * HDIM + ccol]      = acc0[v];
            Out[(size_t)(m0 + 16 + crow + v) * HDIM + ccol] = acc1[v];
        }
    } else {
#pragma unroll
        for (int v = 0; v < 8; ++v) {
            if (m0 + crow + v < N)      Out[(size_t)(m0 + crow + v) * HDIM + ccol] = acc0[v];
            if (m0 + 16 + crow + v < N) Out[(size_t)(m0 + 16 + crow + v) * HDIM + ccol] = acc1[v];
        }
    }
}

// ---------------- graph aggregation ----------------

// self-loop term initializes the accumulator: agg[i] = dinv[i]^2 * h[i]
__global__ void k_agg_init(const float* __restrict__ h, const float* __restrict__ dinv,
                           float* __restrict__ agg, int n4) {
    int i = blockIdx.x * blockDim.x + threadIdx.x;
    int st = gridDim.x * blockDim.x;
    for (; i < n4; i += st) {
        int node = i >> 5;                       // 32 float4 per node row
        float d = dinv[node];
        float w = d * d;
        float4 v = ((const float4*)h)[i];
        v.x *= w; v.y *= w; v.z *= w; v.w *= w;
        ((float4*)agg)[i] = v;
    }
}

// one wave32 per edge: lane moves a float4 of h[src], atomically adds into agg[dst];
// prefetch next edge's source row to overlap the gather with the atomic drain
__global__ __launch_bounds__(256) void k_scatter(const int* __restrict__ src,
                                                 const int* __restrict__ dst,
                                                 const float* __restrict__ dinv,
                                                 const float* __restrict__ h,
                                                 float* __restrict__ agg, int E) {
    int gid  = blockIdx.x * blockDim.x + threadIdx.x;
    int e    = gid >> 5;
    int lane = threadIdx.x & 31;
    int est  = (gridDim.x * blockDim.x) >> 5;
    for (; e < E; e += est) {
        int s = src[e], d = dst[e];
        if (e + est < E) {
            int sn = src[e + est];
            __builtin_prefetch(h + (size_t)sn * HDIM + lane * 4, 0, 3); // global_prefetch
        }
        float w = dinv[s] * dinv[d];
        float4 v = ((const float4*)(h + (size_t)s * HDIM))[lane];
        float* base = agg + (size_t)d * HDIM + lane * 4;
        atomicAdd(base + 0, w * v.x);
        atomicAdd(base + 1, w * v.y);
        atomicAdd(base + 2, w * v.z);
        atomicAdd(base + 3, w * v.w);
    }
}

// out = relu((agg + bias - mean) * rsqrt(var+eps) * gamma + beta)
__global__ void k_bias_bn_relu(const float* __restrict__ agg, const float* __restrict__ bias,
                               const float* __restrict__ gam, const float* __restrict__ bet,
                               const float* __restrict__ mean, const float* __restrict__ var,
                               float* __restrict__ out, int nh) {
    int i = blockIdx.x * blockDim.x + threadIdx.x;
    int st = gridDim.x * blockDim.x;
    for (; i < nh; i += st) {
        int f = i & (HDIM - 1);
        float s = rsqrtf(var[f] + EPSV) * gam[f];
        float v = (agg[i] + bias[f] - mean[f]) * s + bet[f];
        out[i] = fmaxf(v, 0.f);
    }
}

// ---------------- pooling + head ----------------

__global__ __launch_bounds__(256) void k_pool(const int* __restrict__ batch,
                                              const float* __restrict__ h,
                                              float* __restrict__ sums,
                                              float* __restrict__ cnts, int N) {
    int gid  = blockIdx.x * blockDim.x + threadIdx.x;
    int i    = gid >> 5;
    int lane = threadIdx.x & 31;
    int st   = (gridDim.x * blockDim.x) >> 5;
    for (; i < N; i += st) {
        int b = batch[i];
        float4 v = ((const float4*)(h + (size_t)i * HDIM))[lane];
        float* base = sums + (size_t)b * HDIM + lane * 4;
        atomicAdd(base + 0, v.x);
        atomicAdd(base + 1, v.y);
        atomicAdd(base + 2, v.z);
        atomicAdd(base + 3, v.w);
        if (lane == 0) atomicAdd(&cnts[b], 1.0f);
    }
}

// one block (64 threads) per graph: relu(pooled @ fcw1 + fcb1) @ fcw2 + fcb2
__global__ __launch_bounds__(64) void k_head(const float* __restrict__ sums,
                                             const float* __restrict__ cnts,
                                             const float* __restrict__ fcw1,
                                             const float* __restrict__ fcb1,
                                             const float* __restrict__ fcw2,
                                             const float* __restrict__ fcb2,
                                             float* __restrict__ out) {
    __shared__ float pooled[HDIM];
    __shared__ float hid[64];
    int g = blockIdx.x;
    int t = threadIdx.x;
    float inv = 1.0f / fmaxf(cnts[g], 1.0f);
    pooled[t]      = sums[(size_t)g * HDIM + t] * inv;
    pooled[t + 64] = sums[(size_t)g * HDIM + t + 64] * inv;
    __syncthreads();
    float a = fcb1[t];
#pragma unroll 8
    for (int k = 0; k < HDIM; ++k) a += pooled[k] * fcw1[k * 64 + t];
    hid[t] = fmaxf(a, 0.f);
    __syncthreads();
    if (t < CNUM) {
        float o = fcb2[t];
#pragma unroll 8
        for (int j = 0; j < 64; ++j) o += hid[j] * fcw2[j * CNUM + t];
        out[g * CNUM + t] = o;
    }
}

// ---------------- launch ----------------

extern "C" void kernel_launch(void* const* d_in, const int* in_sizes, int n_in,
                              void* d_out, int out_size, void* d_ws, size_t ws_size,
                              hipStream_t stream) {
    const float* x   = (const float*)d_in[0];
    const int* ei    = (const int*)d_in[1];
    const int* batch = (const int*)d_in[2];
    const float* W[3]  = {(const float*)d_in[3], (const float*)d_in[5], (const float*)d_in[7]};
    const float* B[3]  = {(const float*)d_in[4], (const float*)d_in[6], (const float*)d_in[8]};
    const float* Ga[3] = {(const float*)d_in[9],  (const float*)d_in[13], (const float*)d_in[17]};
    const float* Be[3] = {(const float*)d_in[10], (const float*)d_in[14], (const float*)d_in[18]};
    const float* Mn[3] = {(const float*)d_in[11], (const float*)d_in[15], (const float*)d_in[19]};
    const float* Vr[3] = {(const float*)d_in[12], (const float*)d_in[16], (const float*)d_in[20]};
    const float* fcw1 = (const float*)d_in[21];
    const float* fcb1 = (const float*)d_in[22];
    const float* fcw2 = (const float*)d_in[23];
    const float* fcb2 = (const float*)d_in[24];

    const int N = in_sizes[0] / HDIM;            // 100000
    const int E = in_sizes[1] / 2;               // 1600000
    const int* srcp = ei;
    const int* dstp = ei + E;

    float* ws = (float*)d_ws;
    const size_t NH = (size_t)N * HDIM;
    float* bufX = ws;                            // layer input / output (ping)
    float* bufH = ws + NH;                       // X @ W
    float* bufA = ws + 2 * NH;                   // aggregated messages
    float* dinv = ws + 3 * NH;                   // D^{-1/2}
    float* sums = dinv + N;                      // [G, H]
    float* cnts = sums + (size_t)GNUM * HDIM;    // [G]

    const int nh4 = (int)(NH / 4);
    const int nh  = (int)NH;

    // clean input, compute D^{-1/2} with self loops (deg starts at 1)
    k_clean<<<2048, 256, 0, stream>>>(x, bufX, nh4);
    k_fill<<<(N + 255) / 256, 256, 0, stream>>>(dinv, 1.0f, N);
    k_deg_accum<<<2048, 256, 0, stream>>>(dstp, dinv, E);
    k_rsqrt_inplace<<<(N + 255) / 256, 256, 0, stream>>>(dinv, N);

    for (int l = 0; l < 3; ++l) {
        k_gemm_wmma<<<(N + 31) / 32, 256, 0, stream>>>(bufX, W[l], bufH, N);
        k_agg_init<<<2048, 256, 0, stream>>>(bufH, dinv, bufA, nh4);
        k_scatter<<<(E + 7) / 8, 256, 0, stream>>>(srcp, dstp, dinv, bufH, bufA, E);
        k_bias_bn_relu<<<2048, 256, 0, stream>>>(bufA, B[l], Ga[l], Be[l], Mn[l], Vr[l], bufX, nh);
    }

    // global mean pool + MLP head
    k_fill<<<(GNUM * HDIM + GNUM + 255) / 256, 256, 0, stream>>>(sums, 0.f, GNUM * HDIM + GNUM);
    k_pool<<<(N + 7) / 8, 256, 0, stream>>>(batch, bufX, sums, cnts, N);
    k_head<<<GNUM, 64, 0, stream>>>(sums, cnts, fcw1, fcb1, fcw2, fcb2, (float*)d_out);
}